// GAT_5514738008855
// MI455X (gfx1250) — compile-verified
//
#include <hip/hip_runtime.h>
#include <math.h>

// GAT layer on gfx1250: fused flash-softmax attention with fp32 WMMA
// (V_WMMA_F32_16X16X4_F32) + Tensor Data Mover (tensor_load_to_lds) streaming
// of the adjacency matrix, double-buffered in LDS with TENSORcnt waits.
// N=8192, F_IN=512, F_OUT=64.
//
// ws layout: h[8192*64] f32 | f_src[8192] f32 | f_dst[8192] f32  (~2.1 MB)

#define N_NODES 8192
#define F_IN    512
#define F_OUT   64
#define ALPHA     0.2f
#define OUT_SLOPE 0.01f
#define NEG_BIG  -9.0e15f

typedef __attribute__((ext_vector_type(2))) float v2f;
typedef __attribute__((ext_vector_type(8))) float v8f;
typedef __attribute__((ext_vector_type(4))) unsigned int tdm_v4u;
typedef __attribute__((ext_vector_type(8))) int tdm_v8i;
typedef __attribute__((ext_vector_type(4))) int tdm_v4i;

__device__ __forceinline__ v8f wmma_f32_16x16x4(v2f a, v2f b, v8f c) {
  // D = A(16x4 f32) * B(4x16 f32) + C(16x16 f32)
  return __builtin_amdgcn_wmma_f32_16x16x4_f32(
      /*neg_a=*/false, a, /*neg_b=*/false, b,
      /*c_mod=*/(short)0, c, /*reuse_a=*/false, /*reuse_b=*/false);
}

// TDM: DMA a 16-row x 32-col int32 tile (2 KB) of adj into LDS.
// D# per CDNA5 ISA:
//   group0: count=1 | lds_addr | global_addr[56:0] | type=2
//   group1: data_size=4B, tensor_dim0=tensor_dim1=8192, tile=32x16,
//           tensor_dim0_stride=8192 (elements)
// Tracked with TENSORcnt. This toolchain exposes the 6-arg builtin
// (g0 v4u, g1 v8i, g2 v4i, g3 v4i, extra v8i, cpol i32).
__device__ __forceinline__ void tdm_load_adj_tile(unsigned lds_addr,
                                                  const int* gptr) {
  const unsigned long long ga = (unsigned long long)(uintptr_t)gptr;
  tdm_v4u g0;
  g0.x = 1u;                                   // count=1, user descriptor
  g0.y = lds_addr;                             // LDS byte address
  g0.z = (unsigned)ga;                         // global_addr[31:0]
  g0.w = (unsigned)(ga >> 32) | (2u << 30);    // global_addr[56:32] | type=2
  tdm_v8i g1 = {
      (int)0x20000,                            // data_size=2 (4 bytes)
      (int)((N_NODES & 0xFFFF) << 16),         // tensor_dim0[15:0]=8192
      (int)((N_NODES & 0xFFFF) << 16),         // dim0 hi=0 | tensor_dim1 lo=8192
      (int)(32u << 16),                        // dim1 hi=0 | tile_dim0=32
      16,                                      // tile_dim1=16 (tile_dim2=0)
      N_NODES,                                 // tensor_dim0_stride lo32
      0, 0};
  tdm_v4i gz4 = {0, 0, 0, 0};                  // groups 2/3 unused (2D tile)
  tdm_v8i gz8 = {0, 0, 0, 0, 0, 0, 0, 0};
  __builtin_amdgcn_tensor_load_to_lds(g0, g1, gz4, gz4, gz8, /*cpol=*/0);
}

// ---------------------------------------------------------------------------
// Kernel 1: h = X @ W   (8192x512 @ 512x64), one wave per 16x16 tile of h.
// ---------------------------------------------------------------------------
__global__ __launch_bounds__(128) void gat_gemm_h(const float* __restrict__ X,
                                                  const float* __restrict__ W,
                                                  float* __restrict__ h) {
  const int wave = threadIdx.x >> 5;        // 0..3 -> 16-col tile of F_OUT
  const int lane = threadIdx.x & 31;
  const int m = lane & 15;
  const int half = lane >> 4;
  const int rowbase = blockIdx.x * 16;
  const int col = wave * 16 + m;

  const float* __restrict__ xrow = X + (size_t)(rowbase + m) * F_IN;
  v8f acc = {};
  for (int k0 = 0; k0 < F_IN; k0 += 4) {
    const int ka = k0 + 2 * half;
    v2f a;
    a.x = xrow[ka];
    a.y = xrow[ka + 1];
    v2f b;
    b.x = W[(size_t)ka * F_OUT + col];
    b.y = W[(size_t)(ka + 1) * F_OUT + col];
    acc = wmma_f32_16x16x4(a, b, acc);
  }
#pragma unroll
  for (int r = 0; r < 8; ++r)
    h[(size_t)(rowbase + r + 8 * half) * F_OUT + col] = acc[r];
}

// ---------------------------------------------------------------------------
// Kernel 2: f_src[i] = h[i] . a[0:64],  f_dst[i] = h[i] . a[64:128]
// ---------------------------------------------------------------------------
__global__ __launch_bounds__(256) void gat_attn_vec(const float* __restrict__ h,
                                                    const float* __restrict__ a,
                                                    float* __restrict__ fsrc,
                                                    float* __restrict__ fdst) {
  const int lane = threadIdx.x & 31;
  const int row = blockIdx.x * 8 + (threadIdx.x >> 5);
  const float* __restrict__ hr = h + (size_t)row * F_OUT;
  const float h0 = hr[lane];
  const float h1 = hr[lane + 32];
  float s = h0 * a[lane] + h1 * a[lane + 32];
  float d = h0 * a[F_OUT + lane] + h1 * a[F_OUT + lane + 32];
#pragma unroll
  for (int off = 16; off > 0; off >>= 1) {
    s += __shfl_xor(s, off, 32);
    d += __shfl_xor(d, off, 32);
  }
  if (lane == 0) {
    fsrc[row] = s;
    fdst[row] = d;
  }
}

// ---------------------------------------------------------------------------
// Kernel 3: fused masked-softmax(attention) @ h, flash style, split-K.
// Block = 8 waves = one 16-row tile; wave w owns j in [w*1024, (w+1)*1024).
// adj streamed by TDM into double-buffered LDS tiles (16x32 int32).
// Per 16-j sub-chunk: masked leakyrelu scores -> online softmax -> 4 fp32
// WMMA K-steps x 4 N-tiles. Final cross-wave merge (max/sum/rescale) in LDS.
// ---------------------------------------------------------------------------
__global__ __launch_bounds__(256) void gat_flash(const float* __restrict__ h,
                                                 const int* __restrict__ adj,
                                                 const float* __restrict__ fsrc,
                                                 const float* __restrict__ fdst,
                                                 float* __restrict__ out) {
  // 32 KB arena: TDM staging (8 waves x 2 bufs x 2 KB) during the main loop,
  // reused as the 8 x (16x64 f32) reduction buffer afterwards (same per-wave
  // byte ranges, so no cross-wave aliasing).
  __shared__ __align__(16) int arena[8192];
  __shared__ float stats_m[8][16];
  __shared__ float stats_l[8][16];
  __shared__ float Lsh[16];

  const int lane = threadIdx.x & 31;
  const int m = lane & 15;
  const int half = lane >> 4;
  const int wave = __builtin_amdgcn_readfirstlane((int)(threadIdx.x >> 5));
  const int rowbase = blockIdx.x * 16;
  const int grow = rowbase + m;
  const int wjbase = wave * (N_NODES / 8);   // 1024-wide j slice per wave

  const float2* __restrict__ fd2 = (const float2*)fdst;
  const float fs = fsrc[grow];

  const int* gtile0 = adj + (size_t)rowbase * N_NODES + wjbase;
  const unsigned lds_base =
      (unsigned)(uintptr_t)(&arena[0]) + (unsigned)(wave * 2) * 2048u;

  v8f acc0 = {}, acc1 = {}, acc2 = {}, acc3 = {};
  float rm = NEG_BIG;   // running row max (this lane's row m)
  float rl = 0.0f;      // running row sum

  const int NCH = (N_NODES / 8) / 32;   // 32 chunks of 32 columns
  tdm_load_adj_tile(lds_base, gtile0);  // preload chunk 0

  for (int c = 0; c < NCH; ++c) {
    if (c + 1 < NCH) {
      // previous reads of the target buffer are done; issue next DMA, then
      // wait until the current buffer's DMA has landed (in-order TENSORcnt).
      asm volatile("s_wait_dscnt 0x0" ::: "memory");
      tdm_load_adj_tile(lds_base + (unsigned)(((c + 1) & 1) * 2048),
                        gtile0 + (c + 1) * 32);
      asm volatile("s_wait_tensorcnt 0x1" ::: "memory");
    } else {
      asm volatile("s_wait_tensorcnt 0x0" ::: "memory");
    }
    const int* __restrict__ tile = &arena[(wave * 2 + (c & 1)) * 512];

#pragma unroll
    for (int sub = 0; sub < 2; ++sub) {
      const int jb = c * 32 + sub * 16;   // j offset within this wave's slice
      float pv[8];
      float cm = NEG_BIG;
#pragma unroll
      for (int kk = 0; kk < 4; ++kk) {
        const int jl = sub * 16 + 4 * kk + 2 * half;    // col within LDS tile
        const int2 av = *(const int2*)(tile + m * 32 + jl);
        const float2 dv = fd2[(wjbase + jb) / 2 + 2 * kk + half];
        float e0 = fs + dv.x;
        e0 = (e0 > 0.0f) ? e0 : ALPHA * e0;
        e0 = (av.x > 0) ? e0 : NEG_BIG;
        float e1 = fs + dv.y;
        e1 = (e1 > 0.0f) ? e1 : ALPHA * e1;
        e1 = (av.y > 0) ? e1 : NEG_BIG;
        pv[2 * kk] = e0;
        pv[2 * kk + 1] = e1;
        cm = fmaxf(cm, fmaxf(e0, e1));
      }
      cm = fmaxf(cm, __shfl_xor(cm, 16, 32));
      const float nm = fmaxf(rm, cm);
      const float scale = __expf(rm - nm);

      float ps = 0.0f;
#pragma unroll
      for (int i = 0; i < 8; ++i) {
        pv[i] = __expf(pv[i] - nm);
        ps += pv[i];
      }
      ps += __shfl_xor(ps, 16, 32);
      rl = rl * scale + ps;
      rm = nm;

#pragma unroll
      for (int r = 0; r < 8; ++r) {
        const float sr = __shfl(scale, r + 8 * half, 32);
        acc0[r] *= sr;
        acc1[r] *= sr;
        acc2[r] *= sr;
        acc3[r] *= sr;
      }

#pragma unroll
      for (int kk = 0; kk < 4; ++kk) {
        v2f a;
        a.x = pv[2 * kk];
        a.y = pv[2 * kk + 1];
        const int j0 = wjbase + jb + 4 * kk + 2 * half;
        const float* __restrict__ hr0 = h + (size_t)j0 * F_OUT + m;
        const float* __restrict__ hr1 = hr0 + F_OUT;
        v2f b;
        b.x = hr0[0];  b.y = hr1[0];  acc0 = wmma_f32_16x16x4(a, b, acc0);
        b.x = hr0[16]; b.y = hr1[16]; acc1 = wmma_f32_16x16x4(a, b, acc1);
        b.x = hr0[32]; b.y = hr1[32]; acc2 = wmma_f32_16x16x4(a, b, acc2);
        b.x = hr0[48]; b.y = hr1[48]; acc3 = wmma_f32_16x16x4(a, b, acc3);
      }
    }
  }

  // ---- split-K merge across the 8 waves ----
  if (half == 0) {
    stats_m[wave][m] = rm;
    stats_l[wave][m] = rl;
  }
  __syncthreads();

  float M = NEG_BIG;
#pragma unroll
  for (int w = 0; w < 8; ++w) M = fmaxf(M, stats_m[w][m]);
  float L = 0.0f;
#pragma unroll
  for (int w = 0; w < 8; ++w) L += stats_l[w][m] * __expf(stats_m[w][m] - M);
  const float f = __expf(rm - M);   // this wave's rescale factor for row m
  if (half == 0) Lsh[m] = L;        // identical value from every wave

  // scaled partial tiles into the (reused) arena
  float* red = (float*)arena + wave * 1024;
#pragma unroll
  for (int r = 0; r < 8; ++r) {
    const float fr = __shfl(f, r + 8 * half, 32);
    const int e = (r + 8 * half) * F_OUT + m;
    red[e]      = acc0[r] * fr;
    red[e + 16] = acc1[r] * fr;
    red[e + 32] = acc2[r] * fr;
    red[e + 48] = acc3[r] * fr;
  }
  __syncthreads();

  // sum the 8 partials, normalize, leaky_relu(0.01), coalesced float4 store
  const int e0 = (int)threadIdx.x * 4;   // 0..1020
  const float* __restrict__ redall = (const float*)arena;
  float s0 = 0.f, s1 = 0.f, s2 = 0.f, s3 = 0.f;
#pragma unroll
  for (int w = 0; w < 8; ++w) {
    const float4 rv = *(const float4*)(redall + w * 1024 + e0);
    s0 += rv.x;
    s1 += rv.y;
    s2 += rv.z;
    s3 += rv.w;
  }
  const int row = e0 >> 6;
  const float inv = 1.0f / Lsh[row];
  s0 *= inv; s0 = (s0 > 0.f) ? s0 : OUT_SLOPE * s0;
  s1 *= inv; s1 = (s1 > 0.f) ? s1 : OUT_SLOPE * s1;
  s2 *= inv; s2 = (s2 > 0.f) ? s2 : OUT_SLOPE * s2;
  s3 *= inv; s3 = (s3 > 0.f) ? s3 : OUT_SLOPE * s3;
  float4 o;
  o.x = s0; o.y = s1; o.z = s2; o.w = s3;
  *(float4*)(out + (size_t)(rowbase + row) * F_OUT + (e0 & 63)) = o;
}

// ---------------------------------------------------------------------------
extern "C" void kernel_launch(void* const* d_in, const int* in_sizes, int n_in,
                              void* d_out, int out_size, void* d_ws, size_t ws_size,
                              hipStream_t stream) {
  const float* X   = (const float*)d_in[0];   // [8192, 512]
  const int*   adj = (const int*)d_in[1];     // [8192, 8192]
  const float* W   = (const float*)d_in[2];   // [512, 64]
  const float* a   = (const float*)d_in[3];   // [128, 1]
  float* out = (float*)d_out;                 // [1, 8192, 64]

  float* h    = (float*)d_ws;                 // 8192*64
  float* fsrc = h + (size_t)N_NODES * F_OUT;  // 8192
  float* fdst = fsrc + N_NODES;               // 8192

  gat_gemm_h<<<N_NODES / 16, 128, 0, stream>>>(X, W, h);
  gat_attn_vec<<<N_NODES / 8, 256, 0, stream>>>(h, a, fsrc, fdst);
  gat_flash<<<N_NODES / 16, 256, 0, stream>>>(h, adj, fsrc, fdst, out);
}